// GCNEncoder_33552284516498
// MI455X (gfx1250) — compile-verified
//
#include <hip/hip_runtime.h>
#include <hip/hip_bf16.h>

#define N_NODES 50000
#define N_EDGES 800000
#define F_DIM   128
#define H_DIM   128
#define BN_EPS  0.001f

#define A_STRIDE 132   // 16x128 A tile padded: 132*4 = 528 B row stride (16B aligned, bank-spread)

typedef __attribute__((ext_vector_type(2))) float v2f;
typedef __attribute__((ext_vector_type(8))) float v8f;

// ---------------------------------------------------------------------------
// Kernel 0: zero-init the scatter accumulator and degree array
// ---------------------------------------------------------------------------
__global__ void gcn_zero_kernel(float* __restrict__ buf, int n) {
    int i = blockIdx.x * blockDim.x + threadIdx.x;
    if (i < n) buf[i] = 0.0f;
}

// ---------------------------------------------------------------------------
// Kernel 1: per-src degree (float, to mirror the reference's segment_sum of 1s)
// ---------------------------------------------------------------------------
__global__ void gcn_degree_kernel(const int* __restrict__ ep, float* __restrict__ deg, int nE) {
    int e = blockIdx.x * blockDim.x + threadIdx.x;
    if (e < nE) {
        int src = ep[2 * e];
        atomicAdd(&deg[src], 1.0f);
    }
}

// ---------------------------------------------------------------------------
// Kernel 2: t = X @ W + b via V_WMMA_F32_16X16X4_F32, LDS-staged operands.
//
// Block = 256 threads (8 waves) = one 16-row M tile x all 8 N tiles.
//  - W (128x128, 64KB) staged once per block, coalesced b128 -> ds_store_b128.
//    B-fragment reads are ds_load_b32 with lane stride 1 float (conflict-free).
//  - A tile (16x128) staged once, padded stride 132 floats so fragment
//    ds_load_b64 reads hit distinct banks (stride 528B -> bank step 4).
//
// f32 A (16x4):  lanes 0-15: M=lane, K pair {0,1}; lanes 16-31: K pair {2,3}
// f32 B (4x16):  lanes 0-15: N=lane, K={0,1};      lanes 16-31: K={2,3}
// f32 C/D (16x16): VGPR r: lanes 0-15 -> M=r; lanes 16-31 -> M=r+8; N=lane&15
// ---------------------------------------------------------------------------
__global__ void gcn_gemm_wmma_kernel(const float* __restrict__ X,
                                     const float* __restrict__ W,
                                     const float* __restrict__ bias,
                                     float* __restrict__ T) {
    __shared__ float As[16 * A_STRIDE];
    __shared__ float Ws[H_DIM * F_DIM];

    const int tid   = threadIdx.x;
    const int tileM = blockIdx.x * 16;

    // ---- stage A tile: 16x128 = 512 float4, 2 per thread, coalesced ----
    {
        const float4* __restrict__ Xv = (const float4*)(X + (size_t)tileM * F_DIM);
        #pragma unroll
        for (int i = 0; i < 2; ++i) {
            const int j    = tid + i * 256;       // 0..511
            const int row  = j >> 5;              // 32 float4 per row
            const int colf = (j & 31) * 4;
            *(float4*)(&As[row * A_STRIDE + colf]) = Xv[j];
        }
    }
    // ---- stage W: 128x128 = 4096 float4, 16 per thread, coalesced ----
    {
        const float4* __restrict__ Wv = (const float4*)W;
        float4* __restrict__ Wsv = (float4*)Ws;
        #pragma unroll
        for (int i = 0; i < 16; ++i) {
            Wsv[tid + i * 256] = Wv[tid + i * 256];
        }
    }
    __syncthreads();

    const int lane  = tid & 31;
    const int wave  = tid >> 5;              // 0..7 -> N tile
    const int tileN = wave * 16;
    const int half  = lane >> 4;             // which K pair within a chunk
    const int l15   = lane & 15;

    const float* __restrict__ arow = &As[l15 * A_STRIDE];
    const float* __restrict__ bcol = &Ws[tileN + l15];

    v8f c = {};
    #pragma unroll
    for (int kc = 0; kc < F_DIM / 4; ++kc) {
        const int kbase = kc * 4 + half * 2;
        v2f a, b;
        a.x = arow[kbase];
        a.y = arow[kbase + 1];
        b.x = bcol[kbase * H_DIM];
        b.y = bcol[(kbase + 1) * H_DIM];
        // 8 args: (neg_a, A, neg_b, B, c_mod, C, reuse_a, reuse_b)
        c = __builtin_amdgcn_wmma_f32_16x16x4_f32(false, a, false, b,
                                                  (short)0, c, false, false);
    }

    const int col = tileN + l15;
    const float bv = bias[col];
    #pragma unroll
    for (int r = 0; r < 8; ++r) {
        const int row = tileM + half * 8 + r;
        T[(size_t)row * H_DIM + col] = c[r] + bv;
    }
}

// ---------------------------------------------------------------------------
// Kernel 3: edge scatter. One wave per edge; lane l moves cols [4l, 4l+4).
// nb[src] += t[dst] * rsqrt(deg[src]) * rsqrt(deg[dst])
// ---------------------------------------------------------------------------
__global__ void gcn_edge_kernel(const int* __restrict__ ep,
                                const float* __restrict__ T,
                                const float* __restrict__ deg,
                                float* __restrict__ nb, int nE) {
    const int wid  = (blockIdx.x * blockDim.x + threadIdx.x) >> 5;
    const int lane = threadIdx.x & 31;
    if (wid >= nE) return;

    const int src = ep[2 * wid];
    const int dst = ep[2 * wid + 1];
    const float s = __frsqrt_rn(deg[src]) * __frsqrt_rn(deg[dst]);

    const float4* __restrict__ trow = (const float4*)(T + (size_t)dst * H_DIM);
    float* __restrict__ nrow = nb + (size_t)src * H_DIM;

    const float4 v = trow[lane];
    const int c = lane * 4;
    atomicAdd(nrow + c + 0, v.x * s);
    atomicAdd(nrow + c + 1, v.y * s);
    atomicAdd(nrow + c + 2, v.z * s);
    atomicAdd(nrow + c + 3, v.w * s);
}

// ---------------------------------------------------------------------------
// Kernel 4: batch-norm epilogue with analytic src_mean = deg * t
// ---------------------------------------------------------------------------
__global__ void gcn_final_kernel(const float* __restrict__ T,
                                 const float* __restrict__ nb,
                                 const float* __restrict__ deg,
                                 const float* __restrict__ gamma,
                                 const float* __restrict__ beta,
                                 const float* __restrict__ mean,
                                 const float* __restrict__ var,
                                 float* __restrict__ out) {
    const int i = blockIdx.x * blockDim.x + threadIdx.x;
    if (i >= N_NODES * H_DIM) return;
    const int node = i >> 7;
    const int col  = i & (H_DIM - 1);

    const float d  = deg[node];
    const float sm = (d > 0.0f) ? d * T[i] : 0.0f;
    const float agg = 0.5f * nb[i] + 0.5f * sm;
    out[i] = (agg - mean[col]) * __frsqrt_rn(var[col] + BN_EPS) * gamma[col] + beta[col];
}

// ---------------------------------------------------------------------------
extern "C" void kernel_launch(void* const* d_in, const int* in_sizes, int n_in,
                              void* d_out, int out_size, void* d_ws, size_t ws_size,
                              hipStream_t stream) {
    const int*   ep    = (const int*)d_in[0];
    const float* X     = (const float*)d_in[1];
    const float* W     = (const float*)d_in[2];
    const float* bias  = (const float*)d_in[3];
    const float* gamma = (const float*)d_in[4];
    const float* beta  = (const float*)d_in[5];
    const float* mean  = (const float*)d_in[6];
    const float* var   = (const float*)d_in[7];
    float* out = (float*)d_out;

    float* T   = (float*)d_ws;                       // 50000*128 floats
    float* nb  = T  + (size_t)N_NODES * H_DIM;       // 50000*128 floats
    float* deg = nb + (size_t)N_NODES * H_DIM;       // 50000 floats

    const int zero_n = N_NODES * H_DIM + N_NODES;    // nb + deg contiguous
    gcn_zero_kernel<<<(zero_n + 255) / 256, 256, 0, stream>>>(nb, zero_n);

    gcn_degree_kernel<<<(N_EDGES + 255) / 256, 256, 0, stream>>>(ep, deg, N_EDGES);

    gcn_gemm_wmma_kernel<<<N_NODES / 16, 256, 0, stream>>>(X, W, bias, T);

    const long long edge_threads = (long long)N_EDGES * 32;
    gcn_edge_kernel<<<(int)(edge_threads / 256), 256, 0, stream>>>(ep, T, deg, nb, N_EDGES);

    gcn_final_kernel<<<(N_NODES * H_DIM + 255) / 256, 256, 0, stream>>>(
        T, nb, deg, gamma, beta, mean, var, out);
}